// InteractionFusionModule_47880295416411
// MI455X (gfx1250) — compile-verified
//
#include <hip/hip_runtime.h>

// ---------------------------------------------------------------------------
// Graph-transformer layer for MI455X (gfx1250, wave32, WMMA).
//   N=100000 nodes, E=1.6M edges, F=128, H=8 heads, D=16, NC=3
// Pipeline:
//   0) zero-init accumulators (agg, denom, max-keys)
//   1) qkv = x @ Wqkv                        [WMMA f32 16x16x4, 4-tile waves]
//   2) per-edge score + segment-max          (ordered-uint atomicMax)
//   3) per-edge exp: denom += e^x, agg += e^x * v[src]   (fused, atomics)
//   4) h = (agg/denom) @ Wout + bout + x     [WMMA, normalization folded into
//                                             the A-fragment load]
//   5) layernorm per row                     (one wave32 per row, shuffles)
//   6) c1 = relu(h @ Wc1 + bc1)              [WMMA]
//   7) out = c1 @ Wc2 + bc2                  (scalar, NC=3)
// ---------------------------------------------------------------------------

typedef float v2f __attribute__((ext_vector_type(2)));
typedef float v8f __attribute__((ext_vector_type(8)));

// ---- order-preserving float<->uint keys for atomicMax-based segment max ----
__device__ __forceinline__ unsigned f32_key(float f) {
  unsigned u = __float_as_uint(f);
  return (u & 0x80000000u) ? ~u : (u | 0x80000000u);
}
__device__ __forceinline__ float key_f32(unsigned k) {
  return (k & 0x80000000u) ? __uint_as_float(k & 0x7FFFFFFFu)
                           : __uint_as_float(~k);
}

// ---- multi-tile fp32 WMMA: one A fragment feeds NT 16x16 column tiles ------
// A: row-major [nrows x lda], B: row-major [K x ldb]
// A frag (16x4): lane(0..15)=M, VGPR0/1 = K{0,1}; lanes 16..31 -> K{2,3}
// B frag (4x16): lane(0..15)=N, VGPR0/1 = K{0,1}; lanes 16..31 -> K{2,3}
// If SCALE: A element (m,k) is multiplied by invd[k>>4] (per-head 1/denom).
// (k0 % 4 == 0 and the +2 lane-half offset never cross a 16-boundary, so the
//  head index is k0>>4 for both elements of the v2f A fragment.)
template <int K, int NT, bool SCALE>
__device__ __forceinline__ void wmma_mt(const float* __restrict__ A,
                                        const float* __restrict__ B,
                                        int arow_idx, int col0,
                                        int lda, int ldb,
                                        v8f c[NT],
                                        const float* __restrict__ invd) {
  const int lane = threadIdx.x & 31;
  const int m16  = lane & 15;
  const int kh   = (lane >> 4) & 1;
  const float* __restrict__ arow = A + (size_t)arow_idx * lda + 2 * kh;
  const float* __restrict__ brow = B + (size_t)(2 * kh) * ldb + col0 + m16;
#pragma unroll
  for (int k0 = 0; k0 < K; k0 += 4) {
    v2f a;
    a.x = arow[k0];
    a.y = arow[k0 + 1];
    if constexpr (SCALE) {
      const float s = invd[k0 >> 4];
      a.x *= s;
      a.y *= s;
    }
    const float* __restrict__ bp = brow + (size_t)k0 * ldb;
#pragma unroll
    for (int t = 0; t < NT; ++t) {
      v2f b;
      b.x = bp[t * 16];
      b.y = bp[ldb + t * 16];
      c[t] = __builtin_amdgcn_wmma_f32_16x16x4_f32(false, a, false, b,
                                                   (short)0, c[t], false, false);
    }
  }
}

// ---- 0: zero accumulators --------------------------------------------------
__global__ void k_init(float* __restrict__ agg, float* __restrict__ denom,
                       unsigned* __restrict__ mkey, int n) {
  long long i = (long long)blockIdx.x * blockDim.x + threadIdx.x;
  if (i < (long long)n * 128) agg[i] = 0.0f;
  if (i < (long long)n * 8) {
    denom[i] = 0.0f;
    mkey[i]  = 0u;   // below every real key (== encoded -NaN)
  }
}

// ---- 1: qkv = x @ Wqkv  ([n,128] @ [128,384]); wave = 16x64 of output ------
__global__ void k_qkv(const float* __restrict__ x,
                      const float* __restrict__ Wqkv,
                      float* __restrict__ qkv, int n) {
  const int w = blockIdx.x * (blockDim.x >> 5) + (threadIdx.x >> 5);
  const int rowTiles = (n + 15) >> 4;
  if (w >= rowTiles * 6) return;
  const int rt = w / 6, cg = w % 6;               // 6 groups of 64 columns
  const int lane = threadIdx.x & 31;
  const int m16 = lane & 15, kh = (lane >> 4) & 1;
  int ar = rt * 16 + m16;
  if (ar >= n) ar = n - 1;
  v8f c[4] = {};
  wmma_mt<128, 4, false>(x, Wqkv, ar, cg * 64, 128, 384, c, nullptr);
#pragma unroll
  for (int t = 0; t < 4; ++t) {
    const int col = cg * 64 + t * 16 + m16;
#pragma unroll
    for (int r = 0; r < 8; ++r) {
      const int row = rt * 16 + r + 8 * kh;
      if (row < n) qkv[(size_t)row * 384 + col] = c[t][r];
    }
  }
}

// ---- 2: per-edge attention score + segment max -----------------------------
__global__ void k_score(const float* __restrict__ qkv,
                        const int* __restrict__ ei,
                        const float* __restrict__ ea,
                        const float* __restrict__ Wedge,
                        float* __restrict__ score,
                        unsigned* __restrict__ mkey, int E) {
  const long long idx = (long long)blockIdx.x * blockDim.x + threadIdx.x;
  if (idx >= (long long)E * 8) return;
  const int e = (int)(idx >> 3), h = (int)(idx & 7);
  const int src = ei[e];
  const int dst = ei[E + e];
  const float4* __restrict__ q =
      (const float4*)(qkv + (size_t)dst * 384 + h * 16);          // q block
  const float4* __restrict__ k =
      (const float4*)(qkv + (size_t)src * 384 + 128 + h * 16);    // k block
  float s = 0.0f;
#pragma unroll
  for (int t = 0; t < 4; ++t) {
    const float4 a = q[t], b = k[t];
    s += a.x * b.x + a.y * b.y + a.z * b.z + a.w * b.w;
  }
  s = s * 0.25f                                    // D^-0.5, D=16
      + ea[2 * e] * Wedge[h] + ea[2 * e + 1] * Wedge[8 + h];
  score[idx] = s;
  atomicMax(&mkey[(size_t)dst * 8 + h], f32_key(s));
}

// ---- 3: fused exp / denom / unnormalized aggregate -------------------------
//   ex = exp(score - m[dst]); denom[dst,h] += ex; agg[dst,h,:] += ex * v[src]
// Normalization by denom is folded into kernel 4's A-fragment load.
__global__ void k_edgeagg(const float* __restrict__ qkv,
                          const int* __restrict__ ei,
                          const unsigned* __restrict__ mkey,
                          const float* __restrict__ score,
                          float* __restrict__ denom,
                          float* __restrict__ agg, int E) {
  const long long idx = (long long)blockIdx.x * blockDim.x + threadIdx.x;
  if (idx >= (long long)E * 8) return;
  const int e = (int)(idx >> 3), h = (int)(idx & 7);
  const int src = ei[e];
  const int dst = ei[E + e];
  const float* __restrict__ v = qkv + (size_t)src * 384 + 256 + h * 16;
  __builtin_prefetch(v, 0, 3);   // global_prefetch: cover the gather latency
  const float m = key_f32(mkey[(size_t)dst * 8 + h]);
  const float ex = __expf(score[idx] - m);
  atomicAdd(&denom[(size_t)dst * 8 + h], ex);
  float* __restrict__ ag = agg + (size_t)dst * 128 + h * 16;
#pragma unroll
  for (int t = 0; t < 16; ++t) atomicAdd(&ag[t], ex * v[t]);
}

// ---- 4: h = (agg/denom) @ Wout + bout + x ----------------------------------
__global__ void k_outproj(const float* __restrict__ agg,
                          const float* __restrict__ denom,
                          const float* __restrict__ Wout,
                          const float* __restrict__ bout,
                          const float* __restrict__ x,
                          float* __restrict__ hbuf, int n) {
  const int w = blockIdx.x * (blockDim.x >> 5) + (threadIdx.x >> 5);
  const int rowTiles = (n + 15) >> 4;
  if (w >= rowTiles * 2) return;
  const int rt = w >> 1, cg = w & 1;              // 2 groups of 64 columns
  const int lane = threadIdx.x & 31;
  const int m16 = lane & 15, kh = (lane >> 4) & 1;
  int ar = rt * 16 + m16;
  if (ar >= n) ar = n - 1;
  float invd[8];
#pragma unroll
  for (int h = 0; h < 8; ++h)
    invd[h] = 1.0f / (denom[(size_t)ar * 8 + h] + 1e-16f);
  v8f c[4] = {};
  wmma_mt<128, 4, true>(agg, Wout, ar, cg * 64, 128, 128, c, invd);
#pragma unroll
  for (int t = 0; t < 4; ++t) {
    const int col = cg * 64 + t * 16 + m16;
    const float b = bout[col];
#pragma unroll
    for (int r = 0; r < 8; ++r) {
      const int row = rt * 16 + r + 8 * kh;
      if (row < n)
        hbuf[(size_t)row * 128 + col] = c[t][r] + b + x[(size_t)row * 128 + col];
    }
  }
}

// ---- 5: in-place layernorm, one wave32 per row -----------------------------
__global__ void k_layernorm(float* __restrict__ hbuf,
                            const float* __restrict__ lnw,
                            const float* __restrict__ lnb, int n) {
  const int wave = (int)(((long long)blockIdx.x * blockDim.x + threadIdx.x) >> 5);
  const int lane = threadIdx.x & 31;
  if (wave >= n) return;
  float4* __restrict__ row = (float4*)(hbuf + (size_t)wave * 128);
  float4 v = row[lane];
  float s = v.x + v.y + v.z + v.w;
#pragma unroll
  for (int off = 16; off; off >>= 1) s += __shfl_xor(s, off);
  const float mu = s * (1.0f / 128.0f);
  const float4 d = make_float4(v.x - mu, v.y - mu, v.z - mu, v.w - mu);
  float vs = d.x * d.x + d.y * d.y + d.z * d.z + d.w * d.w;
#pragma unroll
  for (int off = 16; off; off >>= 1) vs += __shfl_xor(vs, off);
  const float inv = rsqrtf(vs * (1.0f / 128.0f) + 1e-5f);
  const float4 w = ((const float4*)lnw)[lane];
  const float4 b = ((const float4*)lnb)[lane];
  row[lane] = make_float4(d.x * inv * w.x + b.x, d.y * inv * w.y + b.y,
                          d.z * inv * w.z + b.z, d.w * inv * w.w + b.w);
}

// ---- 6: c1 = relu(h @ Wc1 + bc1)  ([n,128] @ [128,64]) ---------------------
__global__ void k_mlp1(const float* __restrict__ hbuf,
                       const float* __restrict__ Wc1,
                       const float* __restrict__ bc1,
                       float* __restrict__ c1, int n) {
  const int w = blockIdx.x * (blockDim.x >> 5) + (threadIdx.x >> 5);
  const int rowTiles = (n + 15) >> 4;
  if (w >= rowTiles) return;
  const int rt = w;                                // 1 group of 64 columns
  const int lane = threadIdx.x & 31;
  const int m16 = lane & 15, kh = (lane >> 4) & 1;
  int ar = rt * 16 + m16;
  if (ar >= n) ar = n - 1;
  v8f c[4] = {};
  wmma_mt<128, 4, false>(hbuf, Wc1, ar, 0, 128, 64, c, nullptr);
#pragma unroll
  for (int t = 0; t < 4; ++t) {
    const int col = t * 16 + m16;
    const float b = bc1[col];
#pragma unroll
    for (int r = 0; r < 8; ++r) {
      const int row = rt * 16 + r + 8 * kh;
      if (row < n) c1[(size_t)row * 64 + col] = fmaxf(c[t][r] + b, 0.0f);
    }
  }
}

// ---- 7: out = c1 @ Wc2 + bc2  ([n,64] @ [64,3]) ----------------------------
__global__ void k_mlp2(const float* __restrict__ c1,
                       const float* __restrict__ Wc2,
                       const float* __restrict__ bc2,
                       float* __restrict__ out, int n) {
  const int i = blockIdx.x * blockDim.x + threadIdx.x;
  if (i >= n) return;
  const float* __restrict__ row = c1 + (size_t)i * 64;
  float a0 = bc2[0], a1 = bc2[1], a2 = bc2[2];
#pragma unroll
  for (int j = 0; j < 64; ++j) {
    const float v = row[j];
    a0 += v * Wc2[j * 3 + 0];
    a1 += v * Wc2[j * 3 + 1];
    a2 += v * Wc2[j * 3 + 2];
  }
  out[(size_t)i * 3 + 0] = a0;
  out[(size_t)i * 3 + 1] = a1;
  out[(size_t)i * 3 + 2] = a2;
}

// ---------------------------------------------------------------------------
extern "C" void kernel_launch(void* const* d_in, const int* in_sizes, int n_in,
                              void* d_out, int out_size, void* d_ws,
                              size_t ws_size, hipStream_t stream) {
  const float* x     = (const float*)d_in[0];
  const int*   ei    = (const int*)d_in[1];
  const float* ea    = (const float*)d_in[2];
  const float* Wqkv  = (const float*)d_in[3];
  const float* Wedge = (const float*)d_in[4];
  const float* Wout  = (const float*)d_in[5];
  const float* bout  = (const float*)d_in[6];
  const float* lnw   = (const float*)d_in[7];
  const float* lnb   = (const float*)d_in[8];
  const float* Wc1   = (const float*)d_in[9];
  const float* bc1   = (const float*)d_in[10];
  const float* Wc2   = (const float*)d_in[11];
  const float* bc2   = (const float*)d_in[12];
  float* out = (float*)d_out;

  const int n = in_sizes[0] / 128;
  const int E = in_sizes[2] / 2;

  // workspace carve-up (all offsets 16B-aligned)
  char* base = (char*)d_ws;
  size_t off = 0;
  float*    qkv   = (float*)(base + off); off += (size_t)n * 384 * 4;
  float*    agg   = (float*)(base + off); off += (size_t)n * 128 * 4;
  float*    hbuf  = (float*)(base + off); off += (size_t)n * 128 * 4;
  float*    c1    = (float*)(base + off); off += (size_t)n * 64 * 4;
  float*    denom = (float*)(base + off); off += (size_t)n * 8 * 4;
  unsigned* mkey  = (unsigned*)(base + off); off += (size_t)n * 8 * 4;
  float*    score = (float*)(base + off); off += (size_t)E * 8 * 4;
  (void)ws_size; (void)n_in; (void)out_size;

  const int rowTiles = (n + 15) >> 4;
  const long long eh = (long long)E * 8;
  const int ehBlocks = (int)((eh + 255) / 256);

  // 0: zero accumulators (covers n*128 elements)
  {
    long long tot = (long long)n * 128;
    k_init<<<(int)((tot + 255) / 256), 256, 0, stream>>>(agg, denom, mkey, n);
  }
  // 1: qkv GEMM, 8 waves per 256-thread block, each wave = 16x64 of output
  {
    int waves = rowTiles * 6;
    k_qkv<<<(waves + 7) / 8, 256, 0, stream>>>(x, Wqkv, qkv, n);
  }
  // 2-3: edge phase (score+max, then fused exp/denom/aggregate)
  k_score<<<ehBlocks, 256, 0, stream>>>(qkv, ei, ea, Wedge, score, mkey, E);
  k_edgeagg<<<ehBlocks, 256, 0, stream>>>(qkv, ei, mkey, score, denom, agg, E);
  // 4: output projection (+softmax normalization folded in) + residual
  {
    int waves = rowTiles * 2;
    k_outproj<<<(waves + 7) / 8, 256, 0, stream>>>(agg, denom, Wout, bout, x,
                                                   hbuf, n);
  }
  // 5: layernorm (one wave per row, 8 rows per block)
  k_layernorm<<<(n + 7) / 8, 256, 0, stream>>>(hbuf, lnw, lnb, n);
  // 6: MLP hidden
  k_mlp1<<<(rowTiles + 7) / 8, 256, 0, stream>>>(hbuf, Wc1, bc1, c1, n);
  // 7: classifier
  k_mlp2<<<(n + 255) / 256, 256, 0, stream>>>(c1, Wc2, bc2, out, n);
}